// Convolution_Feature_Extractor_26783416058417
// MI455X (gfx1250) — compile-verified
//
#include <hip/hip_runtime.h>
#include <stdint.h>

// 4-wide clang ext vector => global_load_b128 / global_store_b128 / ds_load_b128
typedef __attribute__((ext_vector_type(4))) float f4;
// The async-to-LDS builtin's pointer params are pointer-to-v4i (128-bit vector).
typedef int v4i __attribute__((vector_size(16)));
typedef __attribute__((address_space(1))) v4i* gptr_v4i;
typedef __attribute__((address_space(3))) v4i* lptr_v4i;

#ifndef __has_builtin
#define __has_builtin(x) 0
#endif

#if __has_builtin(__builtin_amdgcn_global_load_async_to_lds_b128)
#define USE_ASYNC_LDS 1
#else
#define USE_ASYNC_LDS 0
#endif

template <int N>
__device__ __forceinline__ void athena_wait_asynccnt() {
#if __has_builtin(__builtin_amdgcn_s_wait_asynccnt)
  __builtin_amdgcn_s_wait_asynccnt(N);
#else
  asm volatile("s_wait_asynccnt %0" ::"n"(N) : "memory");
#endif
}

__device__ __forceinline__ float mask_val(int l, int ai, int ae, int sl) {
  // Matches reference: 1 - (a_end - i)/40 if i < a_end; 1 - (i - a_idx)/40 if i < s_len; else 0
  float lf = (float)l;
  float m1 = 1.0f - ((float)ae - lf) / 40.0f;
  float m2 = 1.0f - (lf - (float)ai) / 40.0f;
  return (l < ae) ? m1 : ((l < sl) ? m2 : 0.0f);
}

// One float4 per thread per tile; tile = 256 float4s (4 KB). Double-buffered
// async-to-LDS pipeline, barrier-free (each wave reads only its own lanes' slots,
// ordered purely by ASYNCcnt: async loads complete in issue order).
__launch_bounds__(256)
__global__ void mask_scale_kernel(const float* __restrict__ data,
                                  const int* __restrict__ aIdx,
                                  const int* __restrict__ aLen,
                                  const int* __restrict__ sLen,
                                  float* __restrict__ out,
                                  int L, int H4, int tiles)
{
  const int tid = threadIdx.x;
  const int stride = gridDim.x;
  const f4* __restrict__ in4 = (const f4*)data;
  f4* __restrict__ out4 = (f4*)out;

#if USE_ASYNC_LDS
  __shared__ f4 buf[2][256];
  {
    int f0 = blockIdx.x * 256 + tid;
    __builtin_amdgcn_global_load_async_to_lds_b128(
        (gptr_v4i)(in4 + f0), (lptr_v4i)&buf[0][tid], 0, 0);
  }
  int parity = 0;
  for (int t = blockIdx.x; t < tiles; t += stride) {
    int tn = t + stride;
    if (tn < tiles) {
      int fn = tn * 256 + tid;
      __builtin_amdgcn_global_load_async_to_lds_b128(
          (gptr_v4i)(in4 + fn), (lptr_v4i)&buf[parity ^ 1][tid], 0, 0);
      athena_wait_asynccnt<1>();   // oldest (current tile) complete
    } else {
      athena_wait_asynccnt<0>();
    }
    f4 v = buf[parity][tid];       // ds_load_b128, own lane's slot -> no barrier needed
    parity ^= 1;

    int f   = t * 256 + tid;       // float4 index
    int row = f / H4;              // H4 = H/4 float4s per (b,l) row
    int b   = row / L;
    int l   = row - b * L;
    int ai  = aIdx[b];
    int ae  = ai + aLen[b];
    int sl  = sLen[b];
    float m = mask_val(l, ai, ae, sl);
    f4 r = v * m;
    __builtin_nontemporal_store(r, out4 + f);  // NT store: don't evict L2-resident input
  }
#else
  for (int t = blockIdx.x; t < tiles; t += stride) {
    int f   = t * 256 + tid;
    f4 v    = in4[f];              // RT load: keep input resident in 192MB L2 across replays
    int row = f / H4;
    int b   = row / L;
    int l   = row - b * L;
    int ai  = aIdx[b];
    int ae  = ai + aLen[b];
    int sl  = sLen[b];
    float m = mask_val(l, ai, ae, sl);
    f4 r = v * m;
    __builtin_nontemporal_store(r, out4 + f);
  }
#endif
}

// Generic scalar tail (never launched for the reference shapes: N % 1024 == 0).
__global__ void mask_scale_tail(const float* __restrict__ data,
                                const int* __restrict__ aIdx,
                                const int* __restrict__ aLen,
                                const int* __restrict__ sLen,
                                float* __restrict__ out,
                                int L, int H, long long start, long long total)
{
  long long e = start + (long long)blockIdx.x * 256 + threadIdx.x;
  if (e < total) {
    long long row = e / H;
    int b = (int)(row / L);
    int l = (int)(row % L);
    int ai = aIdx[b];
    int ae = ai + aLen[b];
    int sl = sLen[b];
    out[e] = data[e] * mask_val(l, ai, ae, sl);
  }
}

extern "C" void kernel_launch(void* const* d_in, const int* in_sizes, int n_in,
                              void* d_out, int out_size, void* d_ws, size_t ws_size,
                              hipStream_t stream) {
  const float* data = (const float*)d_in[0];
  const int*   aIdx = (const int*)d_in[1];
  const int*   aLen = (const int*)d_in[2];
  const int*   sLen = (const int*)d_in[3];
  float* out = (float*)d_out;

  const int B = in_sizes[1];                 // 512
  const long long total = (long long)in_sizes[0];
  const int H = 100;                          // 2*DIM_H per reference
  const int L = (int)(total / B / H);         // 512
  const int H4 = H / 4;                       // 25 float4s per (b,l) row

  const long long N4 = total / 4;             // 6,553,600 float4s
  const int tiles = (int)(N4 / 256);          // 25,600 tiles, exact
  const int grid = tiles < 3200 ? tiles : 3200;  // 8 tiles/block -> deep async pipeline

  mask_scale_kernel<<<grid, 256, 0, stream>>>(data, aIdx, aLen, sLen, out, L, H4, tiles);

  const long long done = (long long)tiles * 1024;  // floats covered by vector kernel
  const long long rem = total - done;
  if (rem > 0) {
    int tgrid = (int)((rem + 255) / 256);
    mask_scale_tail<<<tgrid, 256, 0, stream>>>(data, aIdx, aLen, sLen, out, L, H, done, total);
  }
}